// MultiheadAttention_7017976561954
// MI455X (gfx1250) — compile-verified
//
#include <hip/hip_runtime.h>
#include <hip/hip_bf16.h>

typedef __attribute__((ext_vector_type(2))) float v2f;
typedef __attribute__((ext_vector_type(4))) float v4f;
typedef __attribute__((ext_vector_type(8))) float v8f;

#define B_     4
#define S_     4096
#define D_     1024
#define SCHUNK 32          // s-chunks for the first-stage reduction
#define SPAN   (S_ / SCHUNK)   // 128 rows per chunk

// ---------------------------------------------------------------------------
// Kernel 1: partial column-sums of x over s, b128 non-temporal loads.
// partials[sc][b][d] = sum_{s in chunk sc} x[b, s, d]
// grid = (B_, SCHUNK) = (4, 32) blocks, 256 threads; each thread owns 4
// consecutive d's -> one global_load_b128 per row, fully coalesced 4 KB/row
// per block. x is read exactly once -> non-temporal (don't pollute L2).
// ---------------------------------------------------------------------------
__global__ __launch_bounds__(256) void xsum_partial_kernel(
    const float* __restrict__ x, float* __restrict__ partials) {
  const int b  = blockIdx.x;                 // 0..3
  const int sc = blockIdx.y;                 // 0..31
  const int d  = threadIdx.x * 4;            // 0..1020

  const float* p = x + ((size_t)b * S_ + (size_t)sc * SPAN) * D_ + d;
  v4f acc = {0.0f, 0.0f, 0.0f, 0.0f};
#pragma unroll 8
  for (int s = 0; s < SPAN; ++s) {
    v4f t = __builtin_nontemporal_load((const v4f*)(p + (size_t)s * D_));
    acc += t;
  }
  *(v4f*)(partials + ((size_t)sc * B_ + b) * D_ + d) = acc;
}

// ---------------------------------------------------------------------------
// Kernel 2: combine partials (fixed order -> deterministic) into padded A
// matrix (16 x 1024 row-major). Rows 0..3 = xsum[b], rows 4..15 = 0.
// ---------------------------------------------------------------------------
__global__ __launch_bounds__(256) void xsum_combine_kernel(
    const float* __restrict__ partials, float* __restrict__ A) {
  const int e = blockIdx.x * 256 + threadIdx.x;   // 0..16383
  const int m = e >> 10;                          // row 0..15
  const int d = e & 1023;
  float v = 0.0f;
  if (m < B_) {
#pragma unroll
    for (int sc = 0; sc < SCHUNK; ++sc) {
      v += partials[((size_t)sc * B_ + m) * D_ + d];
    }
  }
  A[e] = v;
}

// ---------------------------------------------------------------------------
// Kernel 3: GEMM  Out(16 x 1024) = A(16 x 1024) * W^T, with W stored as
// (1024 rows x 1024 cols) so B-matrix element (k, n) = W[n*1024 + k].
// One wave (32 threads) per 16-column tile; V_WMMA_F32_16X16X4_F32, K-loop
// of 4. Fragment layouts per CDNA5 ISA 7.12.2:
//   A 16x4 f32: lanes 0-15 -> M=lane, VGPR0/1 = K {k0,k0+1}; lanes 16-31 ->
//               M=lane-16, VGPR0/1 = K {k0+2,k0+3}.
//   B 4x16 f32: mirrored with N across lanes.
//   C/D 16x16 f32: lane<16 -> rows 0-7 in VGPR0-7, lane>=16 -> rows 8-15.
// grid.x = 64 tiles, block = 32 (one full wave, EXEC all ones)
// ---------------------------------------------------------------------------
__global__ __launch_bounds__(32) void gemm16_f32wmma_kernel(
    const float* __restrict__ A, const float* __restrict__ W,
    float* __restrict__ Out) {
  const int lane = threadIdx.x;       // 0..31
  const int half = lane >> 4;         // 0 or 1
  const int l    = lane & 15;
  const int nTile = blockIdx.x;       // 0..63
  const int col   = nTile * 16 + l;

  const float* arow = A + (size_t)l * D_;          // A row m = l
  const float* wrow = W + (size_t)col * D_;        // weight row n = col

  v8f c = {};
#pragma unroll 8
  for (int k0 = 0; k0 < D_; k0 += 4) {
    const int ka = k0 + half * 2;                  // this half-wave's K pair
    v2f a = *(const v2f*)(arow + ka);              // global_load_b64
    v2f b = *(const v2f*)(wrow + ka);
    // D = A*B + C  (f32, round-to-nearest-even, exact f32 GEMM)
    c = __builtin_amdgcn_wmma_f32_16x16x4_f32(
        /*neg_a=*/false, a, /*neg_b=*/false, b,
        /*c_mod=*/(short)0, c, /*reuse_a=*/false, /*reuse_b=*/false);
  }

#pragma unroll
  for (int r = 0; r < 8; ++r) {
    Out[(size_t)(half * 8 + r) * D_ + col] = c[r];
  }
}

// ---------------------------------------------------------------------------
// Kernel 4: broadcast y[b, :] (rows 0..3 of Y) to out[b, s, :] for all s.
// Non-temporal b128 stores; 64 MB streaming write, fully coalesced. Y reads
// (16 KB) stay cached and are effectively broadcast from L2/WGP$.
// ---------------------------------------------------------------------------
__global__ __launch_bounds__(256) void broadcast_kernel(
    const float* __restrict__ Y, float* __restrict__ out) {
  const size_t i  = (size_t)blockIdx.x * 256 + threadIdx.x;  // v4f index
  const int    o4 = (int)(i & 255);                          // 1024/4 = 256
  const int    b  = (int)(i >> 20);                          // / (256*4096)
  const v4f v = ((const v4f*)Y)[(size_t)b * 256 + o4];
  __builtin_nontemporal_store(v, (v4f*)out + i);
}

// ---------------------------------------------------------------------------
// Workspace layout (floats):
//   [0,           131072)  partials  (32 x 4 x 1024)       512 KB
//   [131072,      147456)  A   padded xsum (16 x 1024)      64 KB
//   [147456,      163840)  T = A @ w_v^T   (16 x 1024)      64 KB
//   [163840,      180224)  Y = T @ w_o^T   (16 x 1024)      64 KB
// Total: 704 KB.
// ---------------------------------------------------------------------------
extern "C" void kernel_launch(void* const* d_in, const int* in_sizes, int n_in,
                              void* d_out, int out_size, void* d_ws, size_t ws_size,
                              hipStream_t stream) {
  const float* x     = (const float*)d_in[0];             // (4, 4096, 1024)
  const float* w_qkv = (const float*)d_in[1];             // (3072, 1024)
  const float* w_o   = (const float*)d_in[2];             // (1024, 1024)
  const float* w_v   = w_qkv + (size_t)2 * D_ * D_;       // rows 2048..3071
  float* out = (float*)d_out;

  float* ws       = (float*)d_ws;
  float* partials = ws;
  float* A        = ws + 131072;
  float* T        = ws + 147456;
  float* Y        = ws + 163840;

  // 1) partial sums of x over s (128 blocks x 8 waves, b128 NT reads, 64 MB)
  xsum_partial_kernel<<<dim3(B_, SCHUNK), 256, 0, stream>>>(x, partials);

  // 2) fixed-order combine + zero-pad to 16 x 1024 A matrix
  xsum_combine_kernel<<<64, 256, 0, stream>>>(partials, A);

  // 3) T = A @ w_v^T   (v_wmma_f32_16x16x4_f32)
  gemm16_f32wmma_kernel<<<64, 32, 0, stream>>>(A, w_v, T);

  // 4) Y = T @ w_o^T   (v_wmma_f32_16x16x4_f32)
  gemm16_f32wmma_kernel<<<64, 32, 0, stream>>>(T, w_o, Y);

  // 5) broadcast Y rows 0..3 across all 4096 sequence positions (64 MB NT write)
  const int n4 = (B_ * S_ * D_) / 4;          // 4,194,304 b128 stores
  broadcast_kernel<<<n4 / 256, 256, 0, stream>>>(Y, out);
}